// SampleAndGroup_44324062494987
// MI455X (gfx1250) — compile-verified
//
#include <hip/hip_runtime.h>

#define B_  4
#define N_  16384
#define S_  1024
#define K_  32
#define C_  128
#define RAD2 0.04f

typedef __attribute__((ext_vector_type(2))) float v2f;
typedef __attribute__((ext_vector_type(8))) float v8f;

// ---------------------------------------------------------------------------
// Kernel 1: farthest point sampling. One workgroup per batch, 1024 threads,
// 16 points per thread held in registers (coords + running min-dist).
// 2 barriers per selection round (serial-latency dominated kernel).
// ---------------------------------------------------------------------------
#define FPS_BLK 1024
#define PPT (N_ / FPS_BLK)   // 16

__global__ void fps_kernel(const float* __restrict__ xyz,
                           float* __restrict__ centroids) {
  const int b   = blockIdx.x;
  const int tid = threadIdx.x;
  const float* X = xyz + (size_t)b * N_ * 3;

  float px[PPT], py[PPT], pz[PPT], md[PPT];
#pragma unroll
  for (int i = 0; i < PPT; ++i) {
    int g = tid * PPT + i;
    px[i] = X[3 * g + 0];
    py[i] = X[3 * g + 1];
    pz[i] = X[3 * g + 2];
    md[i] = 1e10f;
  }

  __shared__ float sval[FPS_BLK / 32];
  __shared__ int   sidx[FPS_BLK / 32];
  __shared__ int   s_sel;

  // first selection is index 0
  if (tid == 0) {
    centroids[(size_t)(b * S_) * 3 + 0] = X[0];
    centroids[(size_t)(b * S_) * 3 + 1] = X[1];
    centroids[(size_t)(b * S_) * 3 + 2] = X[2];
  }
  float cx = X[0], cy = X[1], cz = X[2];

  for (int j = 1; j < S_; ++j) {
    float best = -1.0f;
    int   bidx = 0;
#pragma unroll
    for (int i = 0; i < PPT; ++i) {
      float dx = px[i] - cx, dy = py[i] - cy, dz = pz[i] - cz;
      float d = dx * dx + dy * dy + dz * dz;
      md[i] = fminf(md[i], d);
      if (md[i] > best) { best = md[i]; bidx = tid * PPT + i; }
    }
    // wave-level argmax (prefer lower index on ties, matching jnp.argmax)
#pragma unroll
    for (int off = 16; off > 0; off >>= 1) {
      float ov = __shfl_down(best, off, 32);
      int   oi = __shfl_down(bidx, off, 32);
      if (ov > best || (ov == best && oi < bidx)) { best = ov; bidx = oi; }
    }
    if ((tid & 31) == 0) { sval[tid >> 5] = best; sidx[tid >> 5] = bidx; }
    __syncthreads();                      // barrier 1: sval/sidx visible
    if (tid < 32) {
      best = sval[tid];
      bidx = sidx[tid];
#pragma unroll
      for (int off = 16; off > 0; off >>= 1) {
        float ov = __shfl_down(best, off, 32);
        int   oi = __shfl_down(bidx, off, 32);
        if (ov > best || (ov == best && oi < bidx)) { best = ov; bidx = oi; }
      }
      if (tid == 0) {
        s_sel = bidx;
        centroids[(size_t)(b * S_ + j) * 3 + 0] = X[3 * bidx + 0];
        centroids[(size_t)(b * S_ + j) * 3 + 1] = X[3 * bidx + 1];
        centroids[(size_t)(b * S_ + j) * 3 + 2] = X[3 * bidx + 2];
      }
    }
    __syncthreads();                      // barrier 2: s_sel visible
    const int sel = s_sel;
    cx = X[3 * sel + 0];
    cy = X[3 * sel + 1];
    cz = X[3 * sel + 2];
    // no 3rd barrier needed: next round's shared writes are ordered after
    // barrier 2 (sval) / barrier 1 of the next round (s_sel).
  }
}

// ---------------------------------------------------------------------------
// Kernel 2: ball query via V_WMMA_F32_16X16X4_F32, two tiles per iteration.
// One wave per 16 centroids. dist^2 tile = A(16x4) * B(4x16) + 0 with
//   A[m] = (-2cx, -2cy, -2cz, |c|^2),  B[:,n] = (x, y, z, 1)
// and the |x|^2 term folded into the per-column compare threshold, so the
// accumulator is a hoisted zero vector. The two 16-bit row masks of the
// tile pair concatenate into one 32-bit mask in global-index order, so the
// first-K ctz extraction and the uniform early-exit run once per 2 WMMAs.
// ---------------------------------------------------------------------------
__global__ void ballquery_kernel(const float* __restrict__ xyz,
                                 const float* __restrict__ centroids,
                                 int* __restrict__ idxbuf) {
  const int blk   = blockIdx.x;
  const int b     = blk / (S_ / 16);
  const int stile = blk % (S_ / 16);
  const int lane  = threadIdx.x;
  const int m     = lane & 15;

  const float* X = xyz + (size_t)b * N_ * 3;
  const int s = stile * 16 + m;
  const float* Cc = centroids + (size_t)(b * S_ + s) * 3;
  const float cx = Cc[0], cy = Cc[1], cz = Cc[2];
  const float cc = cx * cx + cy * cy + cz * cz;

  v2f a;
  if (lane < 16) { a.x = -2.0f * cx; a.y = -2.0f * cy; }
  else           { a.x = -2.0f * cz; a.y = cc; }

  v8f czero = {};  // loop-invariant accumulator init

  int cnt = 0;
  int first = -1;
  int* myidx = idxbuf + (size_t)(b * S_ + s) * K_;

  for (int t = 0; t < N_ / 32; ++t) {
    // tile pair: points [t*32, t*32+16) and [t*32+16, t*32+32)
    const int n0 = t * 32 + m;
    const int n1 = n0 + 16;
    const float x0 = X[3 * n0 + 0], y0 = X[3 * n0 + 1], z0 = X[3 * n0 + 2];
    const float x1 = X[3 * n1 + 0], y1 = X[3 * n1 + 1], z1 = X[3 * n1 + 2];
    const float thr0 = RAD2 - (x0 * x0 + y0 * y0 + z0 * z0);
    const float thr1 = RAD2 - (x1 * x1 + y1 * y1 + z1 * z1);

    v2f b0, b1;
    if (lane < 16) { b0.x = x0; b0.y = y0; b1.x = x1; b1.y = y1; }
    else           { b0.x = z0; b0.y = 1.0f; b1.x = z1; b1.y = 1.0f; }

    v8f d0 = __builtin_amdgcn_wmma_f32_16x16x4_f32(
        false, a, false, b0, (short)0, czero, false, false);
    v8f d1 = __builtin_amdgcn_wmma_f32_16x16x4_f32(
        false, a, false, b1, (short)0, czero, false, false);

    // per-VGPR ballots: bit L <-> element (v + 8*(L/16), L%16)
    unsigned bal0[8], bal1[8];
#pragma unroll
    for (int v = 0; v < 8; ++v) bal0[v] = __builtin_amdgcn_ballot_w32(d0[v] <= thr0);
#pragma unroll
    for (int v = 0; v < 8; ++v) bal1[v] = __builtin_amdgcn_ballot_w32(d1[v] <= thr1);

    if (lane < 16) {
      unsigned lo = 0, hi = 0;
#pragma unroll
      for (int v = 0; v < 8; ++v) {
        if (m == v)     { lo = bal0[v] & 0xFFFFu; hi = bal1[v] & 0xFFFFu; }
        if (m == v + 8) { lo = bal0[v] >> 16;     hi = bal1[v] >> 16; }
      }
      unsigned mrow = lo | (hi << 16);   // bit i -> global index t*32 + i
      const int base = t * 32;
      while (mrow && cnt < K_) {
        int bit = __builtin_ctz(mrow);
        mrow &= mrow - 1;
        int gi = base + bit;
        if (first < 0) first = gi;
        myidx[cnt++] = gi;
      }
    }
    // uniform early exit once every row has K hits
    bool done = (lane < 16) ? (cnt >= K_) : true;
    if (__builtin_amdgcn_ballot_w32(done) == 0xFFFFFFFFu) break;
  }

  if (lane < 16) {
    if (first < 0) first = 0;  // cannot happen (centroid is its own hit)
    for (; cnt < K_; ++cnt) myidx[cnt] = first;
  }
}

// ---------------------------------------------------------------------------
// Kernel 3: gather (the bandwidth term: ~134 MB -> ~5.8 us at 23.3 TB/s).
// 256-thread blocks; each wave handles one (b,s,k): 32 lanes x float4 =
// the full 512 B feature row, coalesced. Lanes 0..2 also emit centered xyz.
// ---------------------------------------------------------------------------
__global__ void gather_kernel(const float* __restrict__ xyz,
                              const float* __restrict__ features,
                              const float* __restrict__ centroids,
                              const int* __restrict__ idxbuf,
                              float* __restrict__ gxyz,
                              float* __restrict__ gfeat) {
  const int g    = blockIdx.x * 8 + (threadIdx.x >> 5);  // flattened (b,s,k)
  const int lane = threadIdx.x & 31;
  const int k = g & (K_ - 1);
  const int s = (g >> 5) & (S_ - 1);
  const int b = g >> 15;

  const int idx = idxbuf[(size_t)(b * S_ + s) * K_ + k];

  const float4* src = (const float4*)(features + ((size_t)b * N_ + idx) * C_);
  float4* dst = (float4*)(gfeat + (((size_t)(b * S_ + s)) * K_ + k) * C_);
  dst[lane] = src[lane];

  if (lane < 3) {
    gxyz[(((size_t)(b * S_ + s)) * K_ + k) * 3 + lane] =
        xyz[((size_t)b * N_ + idx) * 3 + lane] -
        centroids[(size_t)(b * S_ + s) * 3 + lane];
  }
}

// ---------------------------------------------------------------------------
extern "C" void kernel_launch(void* const* d_in, const int* in_sizes, int n_in,
                              void* d_out, int out_size, void* d_ws, size_t ws_size,
                              hipStream_t stream) {
  const float* xyz      = (const float*)d_in[0];  // [B,N,3]
  const float* features = (const float*)d_in[1];  // [B,N,C]

  float* out       = (float*)d_out;
  float* centroids = out;                                    // B*S*3   = 12288
  float* gxyz      = out + (size_t)B_ * S_ * 3;              // B*S*K*3 = 393216
  float* gfeat     = gxyz + (size_t)B_ * S_ * K_ * 3;        // B*S*K*C

  int* idxbuf = (int*)d_ws;  // B*S*K ints = 512 KB

  fps_kernel<<<B_, FPS_BLK, 0, stream>>>(xyz, centroids);
  ballquery_kernel<<<B_ * (S_ / 16), 32, 0, stream>>>(xyz, centroids, idxbuf);
  gather_kernel<<<(B_ * S_ * K_) / 8, 256, 0, stream>>>(xyz, features, centroids,
                                                        idxbuf, gxyz, gfeat);
}